// VertexAdder_51848845197899
// MI455X (gfx1250) — compile-verified
//
#include <hip/hip_runtime.h>
#include <cstdint>

namespace {
constexpr int B = 2, V = 2048, FEAT = 128, DIM = 3;
constexpr int E = 3 * V;          // 6144 edges
constexpr int F = V + E;          // 8192 final vertices
constexpr int F4 = FEAT / 4;      // 32 float4 per feature row

// d_out layout (flat float offsets, reference return order):
constexpr long long X_OFF = 0;
constexpr long long C_OFF = (long long)B * F * FEAT;          // 2,097,152
constexpr long long A_OFF = C_OFF + (long long)B * F * DIM;   // 2,146,304
constexpr long long P_OFF = A_OFF + (long long)B * F * F;     // 136,364,032
constexpr long long S_OFF = P_OFF + (long long)B * F * F;     // 270,581,760
constexpr long long ZERO_N4 = ((long long)2 * B * F * F) / 4; // 67,108,864 float4
}

// ---- 1) count upper-triangle nonzeros per row of A[0] (one wave32 per row) ----
__global__ void count_rows(const float* __restrict__ A0, int* __restrict__ cnt) {
    int gw   = (blockIdx.x * blockDim.x + threadIdx.x) >> 5;
    int lane = threadIdx.x & 31;
    if (gw >= V) return;
    int i = gw;
    int c = 0;
    for (int jb = i + 1; jb < V; jb += 32) {
        int j = jb + lane;
        if (j < V && A0[(size_t)i * V + j] != 0.0f) c++;
    }
    for (int off = 16; off > 0; off >>= 1) c += __shfl_down(c, off, 32);
    if (lane == 0) cnt[i] = c;
}

// ---- 2) exclusive prefix scan of 2048 row counts (single block, LDS Hillis-Steele) ----
__global__ void scan_rows(const int* __restrict__ cnt, int* __restrict__ rowbase) {
    __shared__ int s[2048];
    int t = threadIdx.x;                 // 1024 threads, 2 elements each
    s[t] = cnt[t];
    s[t + 1024] = cnt[t + 1024];
    __syncthreads();
    for (int off = 1; off < 2048; off <<= 1) {
        int a = (t >= off) ? s[t - off] : 0;
        int b = (t + 1024 >= off) ? s[t + 1024 - off] : 0;
        __syncthreads();
        s[t] += a;
        s[t + 1024] += b;
        __syncthreads();
    }
    rowbase[t]        = s[t] - cnt[t];             // exclusive
    rowbase[t + 1024] = s[t + 1024] - cnt[t + 1024];
}

// ---- 3) compacted edge list in triu row-major order (wave32 ballot ranking) ----
__global__ void fill_edges(const float* __restrict__ A0, const int* __restrict__ rowbase,
                           int* __restrict__ ei, int* __restrict__ ej) {
    int gw   = (blockIdx.x * blockDim.x + threadIdx.x) >> 5;
    int lane = threadIdx.x & 31;
    if (gw >= V) return;
    int i = gw;
    int k = rowbase[i];
    for (int jb = i + 1; jb < V; jb += 32) {
        int j = jb + lane;
        bool p = (j < V) && (A0[(size_t)i * V + j] != 0.0f);
        unsigned long long m = __ballot(p);        // wave32: low 32 bits valid
        if (p) {
            int rank = (int)__popcll(m & ((1ull << lane) - 1ull));
            ei[k + rank] = i;
            ej[k + rank] = j;
        }
        k += (int)__popcll(m);
    }
}

// ---- 4) poly[b, v] = max over rows i of Pid[b, i, v] (coalesced column max) ----
__global__ void poly_max(const float* __restrict__ Pid, float* __restrict__ poly) {
    int t = blockIdx.x * blockDim.x + threadIdx.x;   // over B*V
    if (t >= B * V) return;
    int b = t / V, v = t - b * V;
    const float* p = Pid + (size_t)b * V * V + v;
    float m = p[0];
    for (int i = 1; i < V; i++) m = fmaxf(m, p[(size_t)i * V]);
    poly[t] = m;
}

// ---- 5) x_new / s_new: copy originals + edge midpoints, float4 vectorized ----
__global__ void feat_out(const float4* __restrict__ src, float4* __restrict__ dst,
                         const int* __restrict__ ei, const int* __restrict__ ej,
                         const float* __restrict__ A) {
    int t = blockIdx.x * blockDim.x + threadIdx.x;   // over B*F*F4
    if (t >= B * F * F4) return;
    int b  = t / (F * F4);
    int r  = t - b * (F * F4);
    int v  = r >> 5;            // F4 == 32
    int f4 = r & 31;
    float4 o;
    if (v < V) {
        o = src[((size_t)b * V + v) * F4 + f4];
    } else {
        int e = v - V;
        int vi = ei[e], vj = ej[e];
        float mv = A[(size_t)b * V * V + (size_t)vi * V + vj];
        float4 a = src[((size_t)b * V + vi) * F4 + f4];
        float4 c = src[((size_t)b * V + vj) * F4 + f4];
        o.x = 0.5f * (a.x + c.x) * mv;
        o.y = 0.5f * (a.y + c.y) * mv;
        o.z = 0.5f * (a.z + c.z) * mv;
        o.w = 0.5f * (a.w + c.w) * mv;
    }
    dst[t] = o;
}

// ---- 6) c_new (DIM=3, elementwise) ----
__global__ void dim_out(const float* __restrict__ c_prev, float* __restrict__ dst,
                        const int* __restrict__ ei, const int* __restrict__ ej,
                        const float* __restrict__ A) {
    int t = blockIdx.x * blockDim.x + threadIdx.x;   // over B*F*DIM
    if (t >= B * F * DIM) return;
    int b = t / (F * DIM);
    int r = t - b * (F * DIM);
    int v = r / DIM;
    int d = r - v * DIM;
    float o;
    if (v < V) {
        o = c_prev[((size_t)b * V + v) * DIM + d];
    } else {
        int e = v - V;
        int vi = ei[e], vj = ej[e];
        float mv = A[(size_t)b * V * V + (size_t)vi * V + vj];
        o = 0.5f * (c_prev[((size_t)b * V + vi) * DIM + d] +
                    c_prev[((size_t)b * V + vj) * DIM + d]) * mv;
    }
    dst[t] = o;
}

// ---- 7) bulk zero-fill of A_new + Pid_new (1.07 GB) via CDNA5 async LDS->global B128 stores ----
// LDS tile is zeroed once; each lane then fires ASYNCcnt-tracked 16B stores with no VGPR data
// traffic, drained with s_wait_asynccnt before wave end.
__global__ void zero_async(float* __restrict__ base, long long n4) {
    __shared__ float4 zb[256];
    float4 z;
    z.x = z.y = z.z = z.w = 0.0f;
    zb[threadIdx.x] = z;
    __syncthreads();
    // Generic pointer to an LDS object: low 32 bits are the LDS byte offset (aperture in high bits).
    unsigned int ldsoff = (unsigned int)(uintptr_t)&zb[threadIdx.x];
    float4* dst = (float4*)base;
    long long idx    = (long long)blockIdx.x * blockDim.x + threadIdx.x;
    long long stride = (long long)gridDim.x * blockDim.x;
    for (; idx < n4; idx += stride) {
        unsigned long long ga = (unsigned long long)(uintptr_t)(dst + idx);
        asm volatile("global_store_async_from_lds_b128 %0, %1, off"
                     :: "v"(ga), "v"(ldsoff) : "memory");
    }
    asm volatile("s_wait_asynccnt 0" ::: "memory");
}

// ---- 8) scatter the 4E entries per matrix (collision-free -> plain stores) ----
__global__ void scatter_edges(float* __restrict__ out,
                              const int* __restrict__ ei, const int* __restrict__ ej,
                              const float* __restrict__ A, const float* __restrict__ poly) {
    int t = blockIdx.x * blockDim.x + threadIdx.x;   // over B*E
    if (t >= B * E) return;
    int b = t / E;
    int e = t - b * E;
    int vi = ei[e], vj = ej[e];
    int vv = V + e;
    float mv = A[(size_t)b * V * V + (size_t)vi * V + vj];
    float pm = poly[b * V + vi] * mv;
    float* An = out + A_OFF + (size_t)b * F * F;
    float* Pn = out + P_OFF + (size_t)b * F * F;
    An[(size_t)vi * F + vv] = mv;
    An[(size_t)vv * F + vi] = mv;
    An[(size_t)vv * F + vj] = mv;
    An[(size_t)vj * F + vv] = mv;
    Pn[(size_t)vi * F + vv] = pm;
    Pn[(size_t)vv * F + vi] = pm;
    Pn[(size_t)vv * F + vj] = pm;
    Pn[(size_t)vj * F + vv] = pm;
}

extern "C" void kernel_launch(void* const* d_in, const int* in_sizes, int n_in,
                              void* d_out, int out_size, void* d_ws, size_t ws_size,
                              hipStream_t stream) {
    (void)in_sizes; (void)n_in; (void)out_size; (void)ws_size;
    const float* x_prev = (const float*)d_in[0];
    const float* c_prev = (const float*)d_in[1];
    const float* A      = (const float*)d_in[2];
    const float* Pid    = (const float*)d_in[3];
    const float* s_prev = (const float*)d_in[4];
    float* out = (float*)d_out;

    // Workspace: cnt[V], rowbase[V], ei[E], ej[E], poly[B*V]  (~82 KB)
    int*   cnt     = (int*)d_ws;
    int*   rowbase = cnt + V;
    int*   ei      = rowbase + V;
    int*   ej      = ei + E;
    float* poly    = (float*)(ej + E);

    count_rows  <<<V / 8,            256, 0, stream>>>(A, cnt);          // 1 wave32/row
    scan_rows   <<<1,               1024, 0, stream>>>(cnt, rowbase);
    fill_edges  <<<V / 8,            256, 0, stream>>>(A, rowbase, ei, ej);
    poly_max    <<<(B * V) / 256,    256, 0, stream>>>(Pid, poly);

    feat_out    <<<(B * F * F4) / 256, 256, 0, stream>>>((const float4*)x_prev,
                                                         (float4*)(out + X_OFF), ei, ej, A);
    feat_out    <<<(B * F * F4) / 256, 256, 0, stream>>>((const float4*)s_prev,
                                                         (float4*)(out + S_OFF), ei, ej, A);
    dim_out     <<<(B * F * DIM) / 256, 256, 0, stream>>>(c_prev, out + C_OFF, ei, ej, A);

    // A_new and Pid_new are contiguous: single 268,435,456-float zero region.
    zero_async  <<<16384,            256, 0, stream>>>(out + A_OFF, ZERO_N4);
    scatter_edges<<<(B * E) / 256,   256, 0, stream>>>(out, ei, ej, A, poly);
}